// jslab_kt_2D_25881472925755
// MI455X (gfx1250) — compile-verified
//
#include <hip/hip_runtime.h>
#include <cmath>

// MI455X / gfx1250. Slab-ocean explicit-Euler scan (memory/latency bound,
// ~94 MB traffic -> ~4us HBM floor) + the one real GEMM (pkt2Kt @ K) done
// with V_WMMA_F32_16X16X4_F32.

typedef __attribute__((ext_vector_type(2))) float v2f;
typedef __attribute__((ext_vector_type(8))) float v8f;

constexpr int   NYX   = 128 * 128;   // grid points
constexpr int   NFORC = 360;         // forcing steps
constexpr int   NDTK  = 10;          // K-knots
constexpr float DTS   = 360.0f;      // integration dt

// Gaussian interpolation weight w[row][k] (pre-normalization).
__device__ __forceinline__ float gauss_w(int row, int k) {
  float d = ((float)row * 3600.0f - ((float)k * 129600.0f + 64800.0f)) *
            (1.0f / 64800.0f);
  return expf(-0.5f * d * d);
}

// ---------------------------------------------------------------------------
// Kernel 1: Kt(360x2) = rownorm(W)(360x10) @ exp(pk)(10x2), via WMMA f32
// 16x16x4. One wave per 16-row tile, 3 K-steps (K padded 10 -> 12).
// ---------------------------------------------------------------------------
__global__ __launch_bounds__(32) void kt_wmma(const float* __restrict__ pk,
                                              float* __restrict__ Kt) {
  const int lane = threadIdx.x;      // wave32, all lanes active (EXEC == ~0)
  const int hi   = lane >> 4;        // 0 / 1
  const int lo   = lane & 15;
  const int tile = blockIdx.x;       // 0..22
  const int row  = tile * 16 + lo;
  const int rc   = row < NFORC ? row : NFORC - 1;

  // B operand: K = exp(pk).reshape(10,2), zero-padded to 12x16.
  // ISA layout (4x16 x3 steps): VGPR r, lane -> Krow = s*4 + r + 2*hi, N = lo.
  float bv[3][2];
#pragma unroll
  for (int s = 0; s < 3; ++s)
#pragma unroll
    for (int r = 0; r < 2; ++r) {
      int kk = s * 4 + r + 2 * hi;
      bv[s][r] = (kk < NDTK && lo < 2) ? expf(pk[kk * 2 + lo]) : 0.0f;
    }

  // Row normalization sum.
  float sum = 0.0f;
#pragma unroll
  for (int k = 0; k < NDTK; ++k) sum += gauss_w(rc, k);
  const float inv = 1.0f / sum;

  v8f c{};  // zero accumulator
#pragma unroll
  for (int s = 0; s < 3; ++s) {
    // A operand: 16x4 f32 tile. lane -> M = lo; VGPR j -> K = s*4 + 2*hi + j.
    v2f a;
#pragma unroll
    for (int j = 0; j < 2; ++j) {
      int kc = s * 4 + 2 * hi + j;
      a[j] = (kc < NDTK) ? gauss_w(rc, kc) * inv : 0.0f;
    }
    v2f bb;
    bb[0] = bv[s][0];
    bb[1] = bv[s][1];
    c = __builtin_amdgcn_wmma_f32_16x16x4_f32(false, a, false, bb, (short)0, c,
                                              false, false);
  }

  // D layout: VGPR r, lane -> M = r + 8*hi, N = lo. Only N<2, M-row<360 valid.
#pragma unroll
  for (int r = 0; r < 8; ++r) {
    int orow = tile * 16 + r + 8 * hi;
    if (orow < NFORC && lo < 2) Kt[orow * 2 + lo] = c[r];
  }
}

// ---------------------------------------------------------------------------
// Kernel 2: per-point Euler scan. One thread per grid point; forcing slices
// double-buffered in registers; next+1 slice prefetched (global_prefetch_b8);
// uniform Kt table staged in LDS.
// ---------------------------------------------------------------------------
__global__ __launch_bounds__(64) void slab_scan(
    const float* __restrict__ TAx, const float* __restrict__ TAy,
    const float* __restrict__ fc, const float* __restrict__ Kt,
    float* __restrict__ U, float* __restrict__ V) {
  __shared__ float sKt[NFORC * 2];
  const int tid = threadIdx.x;
  for (int i = tid; i < NFORC * 2; i += 64) sKt[i] = Kt[i];
  __syncthreads();

  const int   p   = blockIdx.x * 64 + tid;
  const float fcp = fc[p];
  float u = 0.0f, v = 0.0f;
  U[p] = 0.0f;  // row 0 of output is the zero initial condition
  V[p] = 0.0f;

  // Register double-buffer of the two active forcing slices.
  float tax_c = TAx[p], tay_c = TAy[p];
  float tax_n = TAx[NYX + p], tay_n = TAy[NYX + p];

  for (int iout = 0; iout < NFORC - 1; ++iout) {  // iout==359 update is
                                                  // dropped by the reference
    // Warm the cache one extra slice ahead.
    int pf = iout + 3 <= NFORC - 1 ? iout + 3 : NFORC - 1;
    __builtin_prefetch(TAx + (size_t)pf * NYX + p, 0, 3);
    __builtin_prefetch(TAy + (size_t)pf * NYX + p, 0, 3);

    const float ktB0 = sKt[(iout + 1) * 2 + 0];  // Kt[itsup]
    const float ktB1 = sKt[(iout + 1) * 2 + 1];
    const int   itb  = iout * 10;

#pragma unroll
    for (int iin = 0; iin < 10; ++iin) {
      const float aa = (float)iin / 10.0f;  // folds to exact f32 constants
      const float om = 1.0f - aa;
      // JAX dynamic gather: negative indices wrap, OOB reads clamp.
      int i0 = itb + iin - 1;
      if (i0 < 0) i0 += NFORC;
      if (i0 > NFORC - 1) i0 = NFORC - 1;
      const float k0   = om * sKt[i0 * 2 + 0] + aa * ktB0;
      const float k1   = om * sKt[i0 * 2 + 1] + aa * ktB1;
      const float taxt = om * tax_c + aa * tax_n;
      const float tayt = om * tay_c + aa * tay_n;
      const float du   = fcp * v + k0 * taxt - k1 * u;
      const float dv   = -fcp * u + k0 * tayt - k1 * v;
      u = fmaf(DTS, du, u);
      v = fmaf(DTS, dv, v);
    }

    U[(size_t)(iout + 1) * NYX + p] = u;
    V[(size_t)(iout + 1) * NYX + p] = v;

    // Rotate buffers; issue next slice's loads early (consumed next iter).
    tax_c = tax_n;
    tay_c = tay_n;
    const int s2 = iout + 2 <= NFORC - 1 ? iout + 2 : NFORC - 1;
    tax_n = TAx[(size_t)s2 * NYX + p];
    tay_n = TAy[(size_t)s2 * NYX + p];
  }
}

// ---------------------------------------------------------------------------
extern "C" void kernel_launch(void* const* d_in, const int* in_sizes, int n_in,
                              void* d_out, int out_size, void* d_ws,
                              size_t ws_size, hipStream_t stream) {
  const float* pk  = (const float*)d_in[0];
  const float* TAx = (const float*)d_in[1];
  const float* TAy = (const float*)d_in[2];
  const float* fc  = (const float*)d_in[3];

  float* Kt = (float*)d_ws;                       // 720 floats of scratch
  float* U  = (float*)d_out;                      // (360,128,128)
  float* V  = U + (size_t)NFORC * NYX;            // (360,128,128)

  kt_wmma<<<23, 32, 0, stream>>>(pk, Kt);         // 23 tiles of 16 rows
  slab_scan<<<NYX / 64, 64, 0, stream>>>(TAx, TAy, fc, Kt, U, V);
}